// AGNN_77378130805345
// MI455X (gfx1250) — compile-verified
//
#include <hip/hip_runtime.h>
#include <hip/hip_bf16.h>

#define kN    100000
#define kF    500
#define kH    32
#define kC    40

typedef __attribute__((ext_vector_type(2))) float v2f;
typedef __attribute__((ext_vector_type(8))) float v8f;

// ---------------- order-preserving float<->uint map for atomic segment-max ----
__device__ __forceinline__ unsigned f2ord(float f) {
    unsigned u = __float_as_uint(f);
    return (u & 0x80000000u) ? ~u : (u | 0x80000000u);
}
__device__ __forceinline__ float ord2f(unsigned u) {
    unsigned v = (u & 0x80000000u) ? (u & 0x7FFFFFFFu) : ~u;
    return __uint_as_float(v);
}

// ---------------- GEMM1: h = relu(x @ W1 + b1), [N,500]x[500,32] -------------
// One wave per 16-row tile; two 16x16 f32 accumulators (32 cols); K in steps of 4
// using V_WMMA_F32_16X16X4_F32.
//   A 16x4 layout: lanes 0-15 -> M=lane, {K=k,k+1}; lanes 16-31 -> {K=k+2,k+3}
//   B 4x16 layout: lanes 0-15 -> N=lane, rows {k,k+1}; lanes 16-31 rows {k+2,k+3}
//   D 16x16: VGPR v -> M=v (lanes 0-15) / M=v+8 (lanes 16-31), N=lane%16
__global__ void gemm1_relu_wmma(const float* __restrict__ x,
                                const float* __restrict__ W1,
                                const float* __restrict__ b1,
                                float* __restrict__ h) {
    const int wave = threadIdx.x >> 5;
    const int lane = threadIdx.x & 31;
    const long long tile = (long long)blockIdx.x * 4 + wave;
    const long long numTiles = kN / 16;           // 6250
    if (tile >= numTiles) return;                 // wave-uniform exit
    const long long row0 = tile * 16;
    const int lm = lane & 15;
    const int hi = lane >> 4;

    v8f acc0 = {}; v8f acc1 = {};
    const float* xrow = x + (row0 + lm) * (long long)kF;

    for (int k = 0; k < kF; k += 4) {
        const int ka = k + hi * 2;
        v2f a; a.x = xrow[ka]; a.y = xrow[ka + 1];        // contiguous b64 load
        const float* wb = W1 + (long long)ka * kH;        // rows ka, ka+1 of W1
        v2f b0; b0.x = wb[lm];      b0.y = wb[kH + lm];
        v2f b1v; b1v.x = wb[16+lm]; b1v.y = wb[kH + 16 + lm];
        acc0 = __builtin_amdgcn_wmma_f32_16x16x4_f32(false, a, false, b0,
                                                     (short)0, acc0, false, false);
        acc1 = __builtin_amdgcn_wmma_f32_16x16x4_f32(false, a, false, b1v,
                                                     (short)0, acc1, false, false);
    }
    const float bia0 = b1[lm], bia1 = b1[16 + lm];
#pragma unroll
    for (int v = 0; v < 8; ++v) {
        const long long r = row0 + v + hi * 8;
        float c0 = acc0[v] + bia0;
        float c1 = acc1[v] + bia1;
        h[r * kH + lm]      = c0 > 0.f ? c0 : 0.f;
        h[r * kH + 16 + lm] = c1 > 0.f ? c1 : 0.f;
    }
}

// ---------------- per-node inverse L2 norm: rn[i] = rsqrt(sum(h[i]^2)+eps) ---
__global__ void norm_kernel(const float* __restrict__ h, float* __restrict__ rn) {
    const int wave = threadIdx.x >> 5;
    const int lane = threadIdx.x & 31;
    const long long node = (long long)blockIdx.x * 8 + wave;
    if (node >= kN) return;
    float v = h[node * kH + lane];                 // coalesced 128B row read
    float sq = v * v;
#pragma unroll
    for (int off = 16; off > 0; off >>= 1) sq += __shfl_xor(sq, off, 32);
    if (lane == 0) rn[node] = rsqrtf(sq + 1e-12f);
}

// ---------------- zero out / init segment-max and segment-sum ----------------
__global__ void init_kernel(float* __restrict__ out, unsigned* __restrict__ m_int,
                            float* __restrict__ ssum) {
    const long long i = (long long)blockIdx.x * blockDim.x + threadIdx.x;
    if (i < (long long)kN * kH) out[i] = 0.f;
    if (i < kN) { m_int[i] = 0u; ssum[i] = 0.f; }  // 0 == ordered(-NaN): safe min
}

// ---------------- pass 1: edge logits + atomic segment max -------------------
__global__ void logits_kernel(const float* __restrict__ h,
                              const float* __restrict__ rn,
                              const long long* __restrict__ src,
                              const long long* __restrict__ dst,
                              const float* __restrict__ beta_p,  // null -> 1.0
                              float* __restrict__ logits,
                              unsigned* __restrict__ m_int,
                              long long E) {
    const int wave = threadIdx.x >> 5;
    const int lane = threadIdx.x & 31;
    const long long e = (long long)blockIdx.x * 8 + wave;
    if (e >= E) return;
    const long long s = src[e], d = dst[e];
    float p = h[s * kH + lane] * h[d * kH + lane];
#pragma unroll
    for (int off = 16; off > 0; off >>= 1) p += __shfl_xor(p, off, 32);
    if (lane == 0) {
        const float beta = beta_p ? *beta_p : 1.0f;
        const float l = beta * rn[s] * rn[d] * p;
        logits[e] = l;
        atomicMax(m_int + d, f2ord(l));            // global_atomic_max_u32
    }
}

// ---------------- pass 2: exp(l - m[dst]) + atomic segment sum ---------------
__global__ void exp_kernel(const long long* __restrict__ dst,
                           const float* __restrict__ logits,
                           const unsigned* __restrict__ m_int,
                           float* __restrict__ elog,
                           float* __restrict__ ssum, long long E) {
    const long long e = (long long)blockIdx.x * blockDim.x + threadIdx.x;
    if (e >= E) return;
    const long long d = dst[e];
    const float ex = expf(logits[e] - ord2f(m_int[d]));
    elog[e] = ex;                                  // in-place over logits
    atomicAdd(ssum + d, ex);                       // global_atomic_add_f32
}

// ---------------- pass 3: out[dst] += alpha * h[src] -------------------------
__global__ void aggr_kernel(const float* __restrict__ h,
                            const long long* __restrict__ src,
                            const long long* __restrict__ dst,
                            const float* __restrict__ elog,
                            const float* __restrict__ ssum,
                            float* __restrict__ out, long long E) {
    const int wave = threadIdx.x >> 5;
    const int lane = threadIdx.x & 31;
    const long long e = (long long)blockIdx.x * 8 + wave;
    if (e >= E) return;
    const long long s = src[e], d = dst[e];
    const float alpha = elog[e] / (ssum[d] + 1e-16f);
    atomicAdd(out + d * kH + lane, alpha * h[s * kH + lane]);
}

// ---------------- GEMM2: out = h @ W2 + b2, [N,32]x[32,40] -------------------
// One wave per 16-row tile; three 16x16 accumulators (cols 0..47, 40..47 masked
// with cndmask multipliers -- EXEC stays all-ones through every WMMA).
__global__ void gemm2_wmma(const float* __restrict__ h,
                           const float* __restrict__ W2,
                           const float* __restrict__ b2,
                           float* __restrict__ out) {
    const int wave = threadIdx.x >> 5;
    const int lane = threadIdx.x & 31;
    const long long tile = (long long)blockIdx.x * 4 + wave;
    if (tile >= kN / 16) return;
    const long long row0 = tile * 16;
    const int lm = lane & 15;
    const int hi = lane >> 4;
    const int c2 = 32 + lm;
    const bool ok2 = c2 < kC;
    const int cc2 = ok2 ? c2 : 0;
    const float msk2 = ok2 ? 1.f : 0.f;

    v8f acc0 = {}, acc1 = {}, acc2 = {};
    const float* xrow = h + (row0 + lm) * (long long)kH;

#pragma unroll
    for (int k = 0; k < kH; k += 4) {
        const int ka = k + hi * 2;
        v2f a; a.x = xrow[ka]; a.y = xrow[ka + 1];
        const float* wb = W2 + (long long)ka * kC;
        v2f b0;  b0.x  = wb[lm];              b0.y  = wb[kC + lm];
        v2f b1v; b1v.x = wb[16 + lm];         b1v.y = wb[kC + 16 + lm];
        v2f b2v; b2v.x = wb[cc2] * msk2;      b2v.y = wb[kC + cc2] * msk2;
        acc0 = __builtin_amdgcn_wmma_f32_16x16x4_f32(false, a, false, b0,
                                                     (short)0, acc0, false, false);
        acc1 = __builtin_amdgcn_wmma_f32_16x16x4_f32(false, a, false, b1v,
                                                     (short)0, acc1, false, false);
        acc2 = __builtin_amdgcn_wmma_f32_16x16x4_f32(false, a, false, b2v,
                                                     (short)0, acc2, false, false);
    }
    const float bi0 = b2[lm], bi1 = b2[16 + lm], bi2 = ok2 ? b2[cc2] : 0.f;
#pragma unroll
    for (int v = 0; v < 8; ++v) {
        const long long r = row0 + v + hi * 8;
        out[r * kC + lm]      = acc0[v] + bi0;
        out[r * kC + 16 + lm] = acc1[v] + bi1;
        if (ok2) out[r * kC + c2] = acc2[v] + bi2;
    }
}

// ---------------- one AGNN propagation ---------------------------------------
static void run_conv(const float* hin, float* hout,
                     const long long* src, const long long* dst,
                     const float* beta, float* rn, unsigned* m_int, float* ssum,
                     float* logi, long long E, hipStream_t stream) {
    norm_kernel  <<<(kN + 7) / 8, 256, 0, stream>>>(hin, rn);
    init_kernel  <<<((long long)kN * kH + 255) / 256, 256, 0, stream>>>(hout, m_int, ssum);
    logits_kernel<<<(unsigned)((E + 7) / 8), 256, 0, stream>>>(hin, rn, src, dst, beta,
                                                               logi, m_int, E);
    exp_kernel   <<<(unsigned)((E + 255) / 256), 256, 0, stream>>>(dst, logi, m_int,
                                                                   logi, ssum, E);
    aggr_kernel  <<<(unsigned)((E + 7) / 8), 256, 0, stream>>>(hin, src, dst, logi,
                                                               ssum, hout, E);
}

extern "C" void kernel_launch(void* const* d_in, const int* in_sizes, int n_in,
                              void* d_out, int out_size, void* d_ws, size_t ws_size,
                              hipStream_t stream) {
    const float*     x     = (const float*)d_in[0];
    const float*     W1    = (const float*)d_in[1];
    const float*     b1    = (const float*)d_in[2];
    const float*     W2    = (const float*)d_in[3];
    const float*     b2    = (const float*)d_in[4];
    const float*     beta2 = (const float*)d_in[5];
    const long long* ei    = (const long long*)d_in[6];
    const long long  E     = in_sizes[6] / 2;
    const long long* srcp  = ei;       // edge_index[0]
    const long long* dstp  = ei + E;   // edge_index[1]

    // carve workspace
    float* ws = (float*)d_ws;
    float*    hA    = ws;               ws += (size_t)kN * kH;
    float*    hB    = ws;               ws += (size_t)kN * kH;
    float*    rn    = ws;               ws += kN;
    unsigned* m_int = (unsigned*)ws;    ws += kN;
    float*    ssum  = ws;               ws += kN;
    float*    logi  = ws;               /* E floats */

    const unsigned gemmGrid = (kN / 16 + 3) / 4;   // 4 waves / block

    // h = relu(x @ W1 + b1)
    gemm1_relu_wmma<<<gemmGrid, 128, 0, stream>>>(x, W1, b1, hA);
    // prop1 (beta = 1.0)
    run_conv(hA, hB, srcp, dstp, nullptr, rn, m_int, ssum, logi, E, stream);
    // prop2 (beta = beta2)
    run_conv(hB, hA, srcp, dstp, beta2, rn, m_int, ssum, logi, E, stream);
    // out = h @ W2 + b2
    gemm2_wmma<<<gemmGrid, 128, 0, stream>>>(hA, W2, b2, (float*)d_out);
}